// GAT_42795054138064
// MI455X (gfx1250) — compile-verified
//
#include <hip/hip_runtime.h>
#include <math.h>

#define N_NODES 50000
#define E_EDGES 800000
#define E_TOT   (E_EDGES + N_NODES)   // self-loops appended
#define F_IN    128
#define HEADS   4
#define HID     32
#define FHID    (HEADS * HID)         // 128
#define NCLS    8
#define LDS_PAD 132                   // 128 + 4: bank-conflict-free column reads

typedef __attribute__((ext_vector_type(2))) float v2f;
typedef __attribute__((ext_vector_type(8))) float v8f;

// ---------------------------------------------------------------------------
// GEMM: C[M x NCOLS] = A[M x K] * B[K x NCOLS], f32, V_WMMA_F32_16X16X4_F32.
// K, NCOLS compile-time -> fully unrolled k-loop, immediate-offset loads,
// back-to-back WMMAs. grid.x = M/16; blockDim = 32*(NCOLS/16); one wave per
// 16x16 tile. 16xK A tile staged to padded LDS via coalesced float4 loads
// (ds_store_b128), A fragments fed from LDS (ds_load_b64, conflict-free),
// B from global (L2-resident, prefetched). No divergence -> EXEC all ones.
// ---------------------------------------------------------------------------
template <int K, int NCOLS>
__global__ void wmma_gemm_f32(const float* __restrict__ A,
                              const float* __restrict__ B,
                              float* __restrict__ C) {
  __shared__ float ldsA[16 * LDS_PAD];   // 8.45 KB of the 320 KB/WGP pool

  const int tid = threadIdx.x;
  const int rowBase = blockIdx.x * 16;

  // cooperative coalesced stage: 16 rows x K cols (K multiple of 4)
  constexpr int KV4 = K >> 2;            // float4s per row
  for (int q = tid; q < 16 * KV4; q += blockDim.x) {
    const int r  = q / KV4;
    const int c4 = q % KV4;
    const float4 v = ((const float4*)(A + (size_t)(rowBase + r) * K))[c4];
    *(float4*)&ldsA[r * LDS_PAD + c4 * 4] = v;   // ds_store_b128
  }

  const int lane = tid & 31;
  const int wave = tid >> 5;
  const int colBase = wave * 16;
  const int lm = lane & 15;      // M (for A) / N (for B,C)
  const int hi = lane >> 4;      // K sub-pair select

  // prefetch this wave's B column stripe (hoisted, unconditional)
  const float* bcol = B + colBase + lm;
#pragma unroll
  for (int kp = 0; kp < K; kp += 32)
    __builtin_prefetch(bcol + (size_t)kp * NCOLS, 0, 1);  // global_prefetch_b8

  __syncthreads();

  v8f acc = {};
  const float* arow = &ldsA[lm * LDS_PAD];
#pragma unroll
  for (int k0 = 0; k0 < K; k0 += 4) {
    const int kb = k0 + 2 * hi;
    v2f a, b;
    a.x = arow[kb];                                   // ds_load, conflict-free
    a.y = arow[kb + 1];
    b.x = bcol[(size_t)kb * NCOLS];
    b.y = bcol[(size_t)(kb + 1) * NCOLS];
    acc = __builtin_amdgcn_wmma_f32_16x16x4_f32(
        /*neg_a=*/false, a, /*neg_b=*/false, b,
        /*c_mod=*/(short)0, acc, /*reuse_a=*/false, /*reuse_b=*/false);
  }
  // D layout: VGPR r -> row (hi ? 8+r : r), col = lm
#pragma unroll
  for (int r = 0; r < 8; ++r)
    C[(size_t)(rowBase + hi * 8 + r) * NCOLS + colBase + lm] = acc[r];
}

// ---------------------------------------------------------------------------
__global__ void fill_f32(float* __restrict__ p, float v, long n) {
  long i = (long)blockIdx.x * blockDim.x + threadIdx.x;
  if (i < n) p[i] = v;
}

__global__ void pad_w2(const float* __restrict__ W2, float* __restrict__ wpad) {
  int i = blockIdx.x * blockDim.x + threadIdx.x;
  if (i >= 128 * 16) return;
  int k = i >> 4, c = i & 15;
  wpad[i] = (c < NCLS) ? W2[k * NCLS + c] : 0.0f;
}

// per-node attention logits: a_s[n,h] = sum_c h[n,h,c]*att_s[h,c]
__global__ void att_coef(const float* __restrict__ h, int ldh,
                         const float* __restrict__ att_s,
                         const float* __restrict__ att_d,
                         float* __restrict__ as_, float* __restrict__ ad_,
                         int Hh, int Cc) {
  int i = blockIdx.x * blockDim.x + threadIdx.x;
  if (i >= N_NODES * Hh) return;
  int node = i / Hh, hd = i % Hh;
  const float* hp = h + (size_t)node * ldh + hd * Cc;
  const float* sp = att_s + hd * Cc;
  const float* dp = att_d + hd * Cc;
  float s = 0.f, d = 0.f;
  for (int c = 0; c < Cc; ++c) {
    float v = hp[c];
    s = fmaf(v, sp[c], s);
    d = fmaf(v, dp[c], d);
  }
  as_[i] = s;
  ad_[i] = d;
}

__device__ __forceinline__ void atomic_max_f32(float* addr, float val) {
  if (val >= 0.0f) atomicMax((int*)addr, __float_as_int(val));
  else             atomicMin((unsigned int*)addr, __float_as_uint(val));
}

__device__ __forceinline__ void edge_endpoints(const int* __restrict__ ei,
                                               long e, int& src, int& dst) {
  if (e < E_EDGES) { src = ei[e]; dst = ei[E_EDGES + e]; }
  else             { src = dst = (int)(e - E_EDGES); }
}

// pass 1: e = leaky_relu(as[src]+ad[dst]); segment max into m[dst]
__global__ void edge_pass1(const int* __restrict__ ei,
                           const float* __restrict__ as_,
                           const float* __restrict__ ad_,
                           float* __restrict__ ebuf, float* __restrict__ m,
                           int Hh) {
  long i = (long)blockIdx.x * blockDim.x + threadIdx.x;
  long total = (long)E_TOT * Hh;
  if (i >= total) return;
  int hd = (int)(i % Hh);
  long e = i / Hh;
  int src, dst;
  edge_endpoints(ei, e, src, dst);
  float v = as_[(size_t)src * Hh + hd] + ad_[(size_t)dst * Hh + hd];
  v = v > 0.0f ? v : 0.2f * v;     // LeakyReLU(0.2)
  ebuf[i] = v;
  atomic_max_f32(&m[(size_t)dst * Hh + hd], v);
}

// pass 2: ex = exp(e - m[dst]); segment sum into den[dst]
__global__ void edge_pass2(const int* __restrict__ ei,
                           float* __restrict__ ebuf,
                           const float* __restrict__ m,
                           float* __restrict__ den, int Hh) {
  long i = (long)blockIdx.x * blockDim.x + threadIdx.x;
  long total = (long)E_TOT * Hh;
  if (i >= total) return;
  int hd = (int)(i % Hh);
  long e = i / Hh;
  int src, dst;
  edge_endpoints(ei, e, src, dst);
  float v = expf(ebuf[i] - m[(size_t)dst * Hh + hd]);
  ebuf[i] = v;
  atomicAdd(&den[(size_t)dst * Hh + hd], v);
}

// pass 3: agg[dst, f..f+3] += h[src, f..f+3] * ex / (den[dst] + 1e-16)
// one float4 gather + 4 L2 atomics per thread (F = Hh*Cc multiple of 4)
__global__ void edge_aggregate(const int* __restrict__ ei,
                               const float* __restrict__ hfeat, int ldh,
                               const float* __restrict__ ebuf,
                               const float* __restrict__ den,
                               float* __restrict__ agg, int ldagg,
                               int Hh, int Cc) {
  const int F4 = (Hh * Cc) >> 2;
  long i = (long)blockIdx.x * blockDim.x + threadIdx.x;
  long total = (long)E_TOT * F4;
  if (i >= total) return;
  int f = (int)(i % F4) * 4;
  long e = i / F4;
  int hd = f / Cc;
  int src, dst;
  edge_endpoints(ei, e, src, dst);
  float alpha = ebuf[e * Hh + hd] / (den[(size_t)dst * Hh + hd] + 1e-16f);
  const float4 hv = *(const float4*)&hfeat[(size_t)src * ldh + f];
  float* ap = &agg[(size_t)dst * ldagg + f];
  atomicAdd(ap + 0, hv.x * alpha);
  atomicAdd(ap + 1, hv.y * alpha);
  atomicAdd(ap + 2, hv.z * alpha);
  atomicAdd(ap + 3, hv.w * alpha);
}

// bias + BatchNorm(eval) + ELU
__global__ void epilogue_bn_elu(const float* __restrict__ agg,
                                const float* __restrict__ bias,
                                const float* __restrict__ g,
                                const float* __restrict__ bb,
                                const float* __restrict__ rm,
                                const float* __restrict__ rv,
                                float* __restrict__ out) {
  long i = (long)blockIdx.x * blockDim.x + threadIdx.x;
  if (i >= (long)N_NODES * FHID) return;
  int f = (int)(i % FHID);
  float y = agg[i] + bias[f];
  y = (y - rm[f]) * rsqrtf(rv[f] + 1e-5f) * g[f] + bb[f];
  out[i] = y > 0.0f ? y : (expf(y) - 1.0f);
}

// out = log_softmax(agg2 + b2)
__global__ void final_out(const float* __restrict__ agg2,
                          const float* __restrict__ b2,
                          float* __restrict__ out) {
  int n = blockIdx.x * blockDim.x + threadIdx.x;
  if (n >= N_NODES) return;
  float v[NCLS];
  float mx = -INFINITY;
  for (int c = 0; c < NCLS; ++c) {
    v[c] = agg2[(size_t)n * NCLS + c] + b2[c];
    mx = fmaxf(mx, v[c]);
  }
  float s = 0.f;
  for (int c = 0; c < NCLS; ++c) s += expf(v[c] - mx);
  float lse = logf(s);
  for (int c = 0; c < NCLS; ++c)
    out[(size_t)n * NCLS + c] = v[c] - mx - lse;
}

// ---------------------------------------------------------------------------
extern "C" void kernel_launch(void* const* d_in, const int* in_sizes, int n_in,
                              void* d_out, int out_size, void* d_ws,
                              size_t ws_size, hipStream_t stream) {
  const float* x   = (const float*)d_in[0];
  const int*   ei  = (const int*)d_in[1];
  const float* W0  = (const float*)d_in[2];
  const float* as0 = (const float*)d_in[3];
  const float* ad0 = (const float*)d_in[4];
  const float* b0  = (const float*)d_in[5];
  const float* g0  = (const float*)d_in[6];
  const float* bb0 = (const float*)d_in[7];
  const float* rm0 = (const float*)d_in[8];
  const float* rv0 = (const float*)d_in[9];
  const float* W1  = (const float*)d_in[10];
  const float* as1 = (const float*)d_in[11];
  const float* ad1 = (const float*)d_in[12];
  const float* b1  = (const float*)d_in[13];
  const float* g1  = (const float*)d_in[14];
  const float* bb1 = (const float*)d_in[15];
  const float* rm1 = (const float*)d_in[16];
  const float* rv1 = (const float*)d_in[17];
  const float* W2  = (const float*)d_in[18];
  const float* as2 = (const float*)d_in[19];
  const float* ad2 = (const float*)d_in[20];
  const float* b2  = (const float*)d_in[21];
  float* out = (float*)d_out;

  // workspace carve-up (floats)
  float* ws    = (float*)d_ws;
  float* hfeat = ws;                                   // N*128 (also L2 h, ld=16)
  float* agg   = hfeat + (size_t)N_NODES * FHID;       // N*128 (also L2 agg, ld=8)
  float* act   = agg   + (size_t)N_NODES * FHID;       // N*128
  float* asb   = act   + (size_t)N_NODES * FHID;       // N*H
  float* adb   = asb   + (size_t)N_NODES * HEADS;      // N*H
  float* mbuf  = adb   + (size_t)N_NODES * HEADS;      // N*H
  float* dbuf  = mbuf  + (size_t)N_NODES * HEADS;      // N*H
  float* ebuf  = dbuf  + (size_t)N_NODES * HEADS;      // E_TOT*H
  float* wpad  = ebuf  + (size_t)E_TOT * HEADS;        // 128*16

  const int BT = 256;
  auto nb = [](long n, int bt) { return (unsigned)((n + bt - 1) / bt); };

  const float* asw[2] = {as0, as1};
  const float* adw[2] = {ad0, ad1};
  const float* bw[2]  = {b0, b1};
  const float* gw[2]  = {g0, g1};
  const float* bbw[2] = {bb0, bb1};
  const float* rmw[2] = {rm0, rm1};
  const float* rvw[2] = {rv0, rv1};

  for (int l = 0; l < 2; ++l) {
    const float* in = (l == 0) ? x : act;
    const float* W  = (l == 0) ? W0 : W1;
    // h = in @ W  (50000x128 @ 128x128), WMMA f32
    wmma_gemm_f32<128, FHID><<<N_NODES / 16, 256, 0, stream>>>(in, W, hfeat);
    att_coef<<<nb((long)N_NODES * HEADS, BT), BT, 0, stream>>>(
        hfeat, FHID, asw[l], adw[l], asb, adb, HEADS, HID);
    fill_f32<<<nb((long)N_NODES * HEADS, BT), BT, 0, stream>>>(
        mbuf, -INFINITY, (long)N_NODES * HEADS);
    fill_f32<<<nb((long)N_NODES * HEADS, BT), BT, 0, stream>>>(
        dbuf, 0.0f, (long)N_NODES * HEADS);
    fill_f32<<<nb((long)N_NODES * FHID, BT), BT, 0, stream>>>(
        agg, 0.0f, (long)N_NODES * FHID);
    edge_pass1<<<nb((long)E_TOT * HEADS, BT), BT, 0, stream>>>(
        ei, asb, adb, ebuf, mbuf, HEADS);
    edge_pass2<<<nb((long)E_TOT * HEADS, BT), BT, 0, stream>>>(
        ei, ebuf, mbuf, dbuf, HEADS);
    edge_aggregate<<<nb((long)E_TOT * (FHID / 4), BT), BT, 0, stream>>>(
        ei, hfeat, FHID, ebuf, dbuf, agg, FHID, HEADS, HID);
    epilogue_bn_elu<<<nb((long)N_NODES * FHID, BT), BT, 0, stream>>>(
        agg, bw[l], gw[l], bbw[l], rmw[l], rvw[l], act);
  }

  // ---- layer 2: heads=1, C=8, concat=False(mean over 1 head == identity) ----
  pad_w2<<<nb(128 * 16, BT), BT, 0, stream>>>(W2, wpad);
  // h2 = act @ Wpad (50000x128 @ 128x16), reuse hfeat with ld=16
  wmma_gemm_f32<128, 16><<<N_NODES / 16, 32, 0, stream>>>(act, wpad, hfeat);
  att_coef<<<nb((long)N_NODES, BT), BT, 0, stream>>>(
      hfeat, 16, as2, ad2, asb, adb, 1, NCLS);
  fill_f32<<<nb((long)N_NODES, BT), BT, 0, stream>>>(mbuf, -INFINITY,
                                                     (long)N_NODES);
  fill_f32<<<nb((long)N_NODES, BT), BT, 0, stream>>>(dbuf, 0.0f, (long)N_NODES);
  fill_f32<<<nb((long)N_NODES * NCLS, BT), BT, 0, stream>>>(
      agg, 0.0f, (long)N_NODES * NCLS);
  edge_pass1<<<nb((long)E_TOT, BT), BT, 0, stream>>>(ei, asb, adb, ebuf, mbuf, 1);
  edge_pass2<<<nb((long)E_TOT, BT), BT, 0, stream>>>(ei, ebuf, mbuf, dbuf, 1);
  edge_aggregate<<<nb((long)E_TOT * (NCLS / 4), BT), BT, 0, stream>>>(
      ei, hfeat, 16, ebuf, dbuf, agg, NCLS, 1, NCLS);
  final_out<<<nb((long)N_NODES, BT), BT, 0, stream>>>(agg, b2, out);
}